// Fastsum_790273983129
// MI455X (gfx1250) — compile-verified
//
#include <hip/hip_runtime.h>
#include <math.h>

// ---------------------------------------------------------------------------
// Sliced fast Gaussian kernel sum (fastsum via 1-D NDFTs), MI455X / gfx1250.
//
// Roofline: ~4 MB traffic (0.2us @ 23.3TB/s) vs ~270M trig evaluations ->
// compute bound on VALU/trans pipes. Strategy:
//   * projection GEMMs (8192x64)@(64x16) -> fp32 WMMA 16x16x4 (precision:
//     phases are multiplied by h up to 512, so low precision is unusable).
//   * adjoint NDFT: symmetry halves it to k=0..512 trig sums (sincos path).
//   * forward NDFT: Chebyshev recurrence, 1 sincos + 512 FMA steps per
//     (target, slice), coefficients staged in LDS (broadcast reads).
// ---------------------------------------------------------------------------

#define DIMK   64
#define KFT    1024
#define HALFK  512
#define PI_F      3.14159265358979323846f
#define TWO_PI_F  6.28318530717958647692f

typedef __attribute__((ext_vector_type(2))) float v2f;
typedef __attribute__((ext_vector_type(8))) float v8f;

// ---------------- init: zero the max-norm accumulator ----------------------
__global__ void fs_init_kernel(unsigned* maxbits) {
    if (threadIdx.x == 0) *maxbits = 0u;
}

// ---------------- max row-norm over x and y (compare sum-of-squares) -------
__global__ void fs_norm_kernel(const float* __restrict__ x,
                               const float* __restrict__ y,
                               int N, int M, unsigned* maxbits) {
    int r = blockIdx.x * blockDim.x + threadIdx.x;
    if (r >= N + M) return;
    const float* p = (r < N) ? (x + (size_t)r * DIMK)
                             : (y + (size_t)(r - N) * DIMK);
    float s = 0.f;
#pragma unroll
    for (int c = 0; c < DIMK; c += 4) {
        float4 v = *(const float4*)(p + c);
        s = fmaf(v.x, v.x, s);
        s = fmaf(v.y, v.y, s);
        s = fmaf(v.z, v.z, s);
        s = fmaf(v.w, v.w, s);
    }
    // positive floats: bit pattern is order-preserving as unsigned
    atomicMax(maxbits, __float_as_uint(s));
}

// ---------------- setup: sf, kft table (log-space), normalized xis ---------
__global__ void fs_setup_kernel(const float* __restrict__ xis,
                                const int* __restrict__ scale,
                                const unsigned* __restrict__ maxbits,
                                float* __restrict__ sf_out,
                                float* __restrict__ kft,
                                float* __restrict__ xin,
                                int P) {
    __shared__ float invn[32];
    int tid = threadIdx.x;

    float maxn = sqrtf(__uint_as_float(*maxbits));
    float sf   = 0.3f / maxn;                 // X_RANGE / max_norm
    float s    = (float)(*scale) * sf;
    float s2   = s * s;
    if (tid == 0) *sf_out = sf;

    if (tid < P) {
        const float* row = xis + tid * DIMK;
        float ss = 0.f;
        for (int c = 0; c < DIMK; ++c) ss = fmaf(row[c], row[c], ss);
        invn[tid] = rsqrtf(ss);
    }
    __syncthreads();

    // kft[t], h = t - 512
    if (tid < KFT) {
        float v = 0.f;
        if (tid != HALFK) {
            float hk   = (float)(tid - HALFK);
            float logk = logf(fabsf(hk));
            float lf = 32.0f * logf(PI_F)
                     + 32.0f * logf(TWO_PI_F * s2)
                     - lgammaf(32.0f)
                     + 63.0f * logk
                     - 2.0f * PI_F * PI_F * s2 * hk * hk;
            v = expf(lf);
        }
        kft[tid] = v;
    }
    // normalized directions (P*DIMK <= 1024 elements)
    if (tid < P * DIMK) {
        int r = tid / DIMK;
        xin[tid] = xis[tid] * invn[r];
    }
}

// ---------------- projections: xp = 2*pi*sf*(x @ xi^T) via fp32 WMMA -------
// One wave computes one 16x16 tile: 16 rows of x (or y) x 16 slices,
// accumulating over K=64 in 16 chunks of 4 with V_WMMA_F32_16X16X4_F32.
__global__ __launch_bounds__(256)
void fs_proj_kernel(const float* __restrict__ x,
                    const float* __restrict__ y,
                    const float* __restrict__ xin,
                    const float* __restrict__ sf_p,
                    float* __restrict__ xp,
                    float* __restrict__ yp,
                    int N, int M, int P) {
    int lane = threadIdx.x & 31;
    int wid  = threadIdx.x >> 5;
    int tile = blockIdx.x * 8 + wid;          // uniform per wave
    int nt = N >> 4, tt = (N + M) >> 4;
    if (tile >= tt) return;                   // wave-uniform exit: EXEC stays full

    const float* src; float* dst; int row0;
    if (tile < nt) { src = x; dst = xp; row0 = tile << 4; }
    else           { src = y; dst = yp; row0 = (tile - nt) << 4; }

    int half = lane >> 4;                     // 0: K pair {0,1}, 1: K pair {2,3}
    int mrow = lane & 15;

    v8f acc = {0.f, 0.f, 0.f, 0.f, 0.f, 0.f, 0.f, 0.f};
#pragma unroll
    for (int k0 = 0; k0 < DIMK; k0 += 4) {
        // A (16x4 f32): lane<16 -> K=k0..k0+1, lane>=16 -> K=k0+2..k0+3, M=mrow
        const float* ap = src + (size_t)(row0 + mrow) * DIMK + k0 + 2 * half;
        v2f a; a.x = ap[0]; a.y = ap[1];
        // B (4x16 f32): B[k][n] = xin[n][k]; K = k0 + 2*half + vgpr, N = mrow
        const float* bp = xin + mrow * DIMK + k0 + 2 * half;
        v2f b; b.x = bp[0]; b.y = bp[1];
        acc = __builtin_amdgcn_wmma_f32_16x16x4_f32(
            /*neg_a=*/false, a, /*neg_b=*/false, b,
            /*c_mod=*/(short)0, acc, /*reuse_a=*/false, /*reuse_b=*/false);
    }

    float sc = TWO_PI_F * (*sf_p);
    // D layout: VGPR v, lane -> row = v + 8*half, col = mrow
#pragma unroll
    for (int v = 0; v < 8; ++v)
        dst[(size_t)(row0 + v + 8 * half) * P + mrow] = sc * acc[v];
}

// ---------------- adjoint NDFT: S_c[k], S_s[k] for k = 1..512 --------------
// a_re[512+k] =  kft[512+k]*S_c[k]   a_im[512+k] =  kft[512+k]*S_s[k]
// a_re[512-k] =  kft[512-k]*S_c[k]   a_im[512-k] = -kft[512-k]*S_s[k]
__global__ __launch_bounds__(256)
void fs_adjoint_kernel(const float* __restrict__ xp,
                       const float* __restrict__ xw,
                       const float* __restrict__ kft,
                       float* __restrict__ are,
                       float* __restrict__ aim,
                       int N, int P) {
    __shared__ float xpc[256];
    __shared__ float wc[256];
    int tid  = threadIdx.x;
    int p    = blockIdx.x >> 1;
    int kblk = blockIdx.x & 1;
    int k    = kblk * 256 + tid + 1;          // 1..512
    float hf = (float)k;

    float sc = 0.f, ss = 0.f;
    for (int i0 = 0; i0 < N; i0 += 256) {
        __syncthreads();
        xpc[tid] = xp[(size_t)(i0 + tid) * P + p];
        wc[tid]  = xw[i0 + tid];
        __syncthreads();
#pragma unroll 4
        for (int t = 0; t < 256; ++t) {
            float sv, cv;
            __sincosf(hf * xpc[t], &sv, &cv);
            sc = fmaf(wc[t], cv, sc);
            ss = fmaf(wc[t], sv, ss);
        }
    }

    float* arp = are + (size_t)p * KFT;
    float* aip = aim + (size_t)p * KFT;
    if (k <= HALFK - 1) {                     // +k side, h = k
        arp[HALFK + k] = kft[HALFK + k] * sc;
        aip[HALFK + k] = kft[HALFK + k] * ss;
    }
    arp[HALFK - k] =  kft[HALFK - k] * sc;    // -k side, h = -k
    aip[HALFK - k] = -kft[HALFK - k] * ss;
    if (k == 1) { arp[HALFK] = 0.f; aip[HALFK] = 0.f; }  // h = 0 (kft=0)
}

// ---------------- forward NDFT via Chebyshev recurrence --------------------
__global__ __launch_bounds__(256)
void fs_forward_kernel(const float* __restrict__ yp,
                       const float* __restrict__ are,
                       const float* __restrict__ aim,
                       float* __restrict__ out,
                       int M, int P) {
    __shared__ float lre[KFT];
    __shared__ float lim[KFT];
    int tid = threadIdx.x;
    int j   = blockIdx.x * 256 + tid;

    float acc = 0.f;
    for (int p = 0; p < P; ++p) {
        __syncthreads();
        for (int t = tid; t < KFT; t += 256) {
            lre[t] = are[(size_t)p * KFT + t];
            lim[t] = aim[(size_t)p * KFT + t];
        }
        __syncthreads();

        float th = (j < M) ? yp[(size_t)j * P + p] : 0.f;
        float s1, c1;
        __sincosf(th, &s1, &c1);
        float twoc = 2.f * c1;
        float ck = c1, sk = s1, ckm = 1.f, skm = 0.f;

        float a = lre[HALFK];                 // k=0 term (zero by construction)
        for (int k = 1; k < HALFK; ++k) {
            a = fmaf(lre[HALFK + k] + lre[HALFK - k], ck, a);
            a = fmaf(lim[HALFK + k] - lim[HALFK - k], sk, a);
            float cn = fmaf(twoc, ck, -ckm);
            float sn = fmaf(twoc, sk, -skm);
            ckm = ck; ck = cn;
            skm = sk; sk = sn;
        }
        // k = 512 only has the h = -512 coefficient (index 0)
        a = fmaf(lre[0], ck, a);
        a = fmaf(-lim[0], sk, a);
        acc += a;
    }
    if (j < M) out[j] = acc / (float)P;
}

// ---------------------------------------------------------------------------
extern "C" void kernel_launch(void* const* d_in, const int* in_sizes, int n_in,
                              void* d_out, int out_size, void* d_ws, size_t ws_size,
                              hipStream_t stream) {
    const float* x    = (const float*)d_in[0];
    const float* y    = (const float*)d_in[1];
    const float* xw   = (const float*)d_in[2];
    const float* xis  = (const float*)d_in[3];
    const int*   scale = (const int*)d_in[4];

    int N = in_sizes[0] / DIMK;
    int M = in_sizes[1] / DIMK;
    int P = in_sizes[3] / DIMK;

    float* ws = (float*)d_ws;
    unsigned* maxbits = (unsigned*)ws;            // [0]
    float* sfp = ws + 1;                          // [1]
    float* kft = ws + 16;                         // [16 .. 16+1024)
    float* xin = kft + KFT;                       // P*DIMK
    float* xp  = xin + P * DIMK;                  // N*P
    float* yp  = xp + (size_t)N * P;              // M*P
    float* are = yp + (size_t)M * P;              // P*KFT
    float* aim = are + (size_t)P * KFT;           // P*KFT
    float* out = (float*)d_out;

    fs_init_kernel<<<1, 32, 0, stream>>>(maxbits);
    fs_norm_kernel<<<(N + M + 255) / 256, 256, 0, stream>>>(x, y, N, M, maxbits);
    fs_setup_kernel<<<1, 1024, 0, stream>>>(xis, scale, maxbits, sfp, kft, xin, P);

    int tiles = (N + M) / 16;
    fs_proj_kernel<<<(tiles + 7) / 8, 256, 0, stream>>>(x, y, xin, sfp, xp, yp, N, M, P);

    fs_adjoint_kernel<<<P * 2, 256, 0, stream>>>(xp, xw, kft, are, aim, N, P);
    fs_forward_kernel<<<(M + 255) / 256, 256, 0, stream>>>(yp, are, aim, out, M, P);
}